// Temporal_Attention_69432441307597
// MI455X (gfx1250) — compile-verified
//
#include <hip/hip_runtime.h>
#include <hip/hip_bf16.h>

// B=8, T=4096, H=1024.
// Reassociation: score[b,t] = value[b,t,:] . (last[b] @ W); the bias term is a
// per-batch constant over t and is exactly cancelled by softmax shift-invariance.

#define B_  8
#define T_  4096
#define H_  1024

typedef float v2f __attribute__((ext_vector_type(2)));
typedef float v8f __attribute__((ext_vector_type(8)));

// ---------------------------------------------------------------------------
// Kernel 1: q[b, n] = sum_o last[b, o] * W[o, n]   via V_WMMA_F32_16X16X4_F32
// One wave per 16-column tile of q. M dimension = 16 (8 valid batches + 8 pad).
// A (16x4 f32) lane layout: lane L -> row m = L&15; VGPR0 = K=2*(L>>4),
// VGPR1 = K=2*(L>>4)+1.  B (4x16) mirrored: lane L -> col n = L&15, rows
// K=2*(L>>4), 2*(L>>4)+1.  D (16x16 f32): lanes 0..15 hold M=0..7 in c[0..7].
//
// Padding rows (m >= 8) read an aliased in-bounds row (m & 7) unconditionally
// (single global_load_b64) and are zeroed with a branchless v_cndmask select —
// avoids the s_and_saveexec branch blocks the predicated-load version produced.
// ---------------------------------------------------------------------------
__global__ __launch_bounds__(32) void qproj_wmma(const float* __restrict__ last,
                                                 const float* __restrict__ W,
                                                 float* __restrict__ q) {
    const int lane  = threadIdx.x & 31;
    const int m     = lane & 15;       // A-matrix row (batch index, pad >= 8)
    const int half  = lane >> 4;       // selects K pair {0,1} or {2,3}
    const int ncol  = blockIdx.x * 16 + (lane & 15);   // B/D column
    const bool live = (m < B_);

    const float* lrow = last + (m & 7) * H_;   // always in-bounds

    v8f acc = {};
    for (int k = 0; k < H_; k += 4) {
        const int ka = k + 2 * half;

        v2f a = *(const v2f*)(lrow + ka);      // global_load_b64
        a.x = live ? a.x : 0.0f;               // v_cndmask, no exec games
        a.y = live ? a.y : 0.0f;

        v2f bv;
        bv.x = W[(size_t)(ka)     * H_ + ncol];
        bv.y = W[(size_t)(ka + 1) * H_ + ncol];

        acc = __builtin_amdgcn_wmma_f32_16x16x4_f32(
            /*neg_a=*/false, a, /*neg_b=*/false, bv,
            /*c_mod=*/(short)0, acc, /*reuse_a=*/false, /*reuse_b=*/false);
    }

    if (lane < 16) {
        #pragma unroll
        for (int r = 0; r < B_; ++r)
            q[r * H_ + blockIdx.x * 16 + lane] = acc[r];
    }
}

// ---------------------------------------------------------------------------
// Kernel 2: scores[b,t] = value[b,t,:] . q[b,:]
// 4096 blocks x 256 threads; 512 blocks per batch, 8 t-rows per block
// (one per wave). q[b] staged through LDS; float4 coalesced value loads.
// ---------------------------------------------------------------------------
__global__ __launch_bounds__(256) void score_k(const float* __restrict__ value,
                                               const float* __restrict__ q,
                                               float* __restrict__ scores) {
    __shared__ float qs[H_];
    const int b     = blockIdx.x >> 9;          // 512 blocks / batch
    const int tbase = (blockIdx.x & 511) * 8;
    const int tid   = threadIdx.x;

    // cooperative load of q[b] (4 KB) into LDS: 256 threads x float4
    ((float4*)qs)[tid] = ((const float4*)(q + b * H_))[tid];
    __syncthreads();

    const int wave = tid >> 5;
    const int lane = tid & 31;
    const int t    = tbase + wave;

    const float4* v4  = (const float4*)(value + ((size_t)b * T_ + t) * H_);
    const float4* q4  = (const float4*)qs;

    float sum = 0.0f;
    #pragma unroll
    for (int j = 0; j < 8; ++j) {
        const float4 v  = v4[lane + 32 * j];
        const float4 qq = q4[lane + 32 * j];
        sum += v.x * qq.x + v.y * qq.y + v.z * qq.z + v.w * qq.w;
    }

    // wave32 butterfly reduction
    #pragma unroll
    for (int off = 16; off > 0; off >>= 1)
        sum += __shfl_xor(sum, off, 32);

    if (lane == 0)
        scores[b * T_ + t] = sum;
}

// ---------------------------------------------------------------------------
// Kernel 3: attn[b,:] = softmax(scores[b,:]) over T.  One block per batch.
// Fixed-order LDS tree reductions -> deterministic.
// ---------------------------------------------------------------------------
__global__ __launch_bounds__(256) void softmax_k(const float* __restrict__ scores,
                                                 float* __restrict__ attn) {
    __shared__ float red[256];
    const int b   = blockIdx.x;
    const int tid = threadIdx.x;

    float s[16];
    float mx = -3.402823466e38f;
    #pragma unroll
    for (int i = 0; i < 16; ++i) {
        s[i] = scores[b * T_ + i * 256 + tid];   // coalesced
        mx = fmaxf(mx, s[i]);
    }

    red[tid] = mx;
    __syncthreads();
    for (int off = 128; off > 0; off >>= 1) {
        if (tid < off) red[tid] = fmaxf(red[tid], red[tid + off]);
        __syncthreads();
    }
    mx = red[0];
    __syncthreads();

    float e[16];
    float lsum = 0.0f;
    #pragma unroll
    for (int i = 0; i < 16; ++i) {
        e[i] = __expf(s[i] - mx);
        lsum += e[i];
    }

    red[tid] = lsum;
    __syncthreads();
    for (int off = 128; off > 0; off >>= 1) {
        if (tid < off) red[tid] += red[tid + off];
        __syncthreads();
    }
    const float inv = 1.0f / red[0];

    #pragma unroll
    for (int i = 0; i < 16; ++i)
        attn[b * T_ + i * 256 + tid] = e[i] * inv;
}

// ---------------------------------------------------------------------------
// Kernel 4a: partial[b, chunk, h] = sum_{t in chunk (64)} attn[b,t]*value[b,t,h]
// Grid (64 chunks, 8 batches) x 256 threads; each thread owns 4 h's.
// Fully coalesced column accumulation; value should hit L2 (128 MB < 192 MB).
// ---------------------------------------------------------------------------
__global__ __launch_bounds__(256) void ctx_partial(const float* __restrict__ value,
                                                   const float* __restrict__ attn,
                                                   float* __restrict__ part) {
    const int ch  = blockIdx.x;    // 0..63
    const int b   = blockIdx.y;    // 0..7
    const int tid = threadIdx.x;

    const float* vbase = value + ((size_t)b * T_ + ch * 64) * H_;
    const float* abase = attn + b * T_ + ch * 64;

    float a0 = 0.f, a1 = 0.f, a2 = 0.f, a3 = 0.f;
    for (int tt = 0; tt < 64; ++tt) {
        const float  w  = abase[tt];
        const float* vr = vbase + (size_t)tt * H_;
        a0 += w * vr[tid];
        a1 += w * vr[tid + 256];
        a2 += w * vr[tid + 512];
        a3 += w * vr[tid + 768];
    }

    float* p = part + ((size_t)(b * 64 + ch)) * H_;
    p[tid]       = a0;
    p[tid + 256] = a1;
    p[tid + 512] = a2;
    p[tid + 768] = a3;
}

// ---------------------------------------------------------------------------
// Kernel 4b: out[b,0,h] = sum_{chunk=0..63} partial[b,chunk,h]
// ---------------------------------------------------------------------------
__global__ __launch_bounds__(256) void ctx_reduce(const float* __restrict__ part,
                                                  float* __restrict__ out) {
    const int idx = blockIdx.x * 256 + threadIdx.x;   // 0..8191
    const int b   = idx >> 10;
    const int h   = idx & (H_ - 1);

    const float* p = part + (size_t)b * 64 * H_ + h;
    float s = 0.0f;
    #pragma unroll
    for (int c = 0; c < 64; ++c)
        s += p[c * H_];
    out[idx] = s;
}

// ---------------------------------------------------------------------------
extern "C" void kernel_launch(void* const* d_in, const int* in_sizes, int n_in,
                              void* d_out, int out_size, void* d_ws, size_t ws_size,
                              hipStream_t stream) {
    const float* value = (const float*)d_in[0];   // [B, T, H]
    const float* last  = (const float*)d_in[1];   // [B, H]
    const float* W     = (const float*)d_in[2];   // [H, H]
    // d_in[3] = bias [H]: provably cancelled by softmax shift-invariance.
    float* out = (float*)d_out;                   // [B, 1, H] -> 8192 floats

    char*  ws     = (char*)d_ws;
    float* q      = (float*)(ws);                 //  32 KB: [B, H]
    float* scores = (float*)(ws + (32  << 10));   // 128 KB: [B, T]
    float* attn   = (float*)(ws + (160 << 10));   // 128 KB: [B, T]
    float* part   = (float*)(ws + (512 << 10));   //   2 MB: [B, 64, H]

    qproj_wmma <<<64,            32, 0, stream>>>(last, W, q);
    score_k    <<<4096,         256, 0, stream>>>(value, q, scores);
    softmax_k  <<<B_,           256, 0, stream>>>(scores, attn);
    ctx_partial<<<dim3(64, B_), 256, 0, stream>>>(value, attn, part);
    ctx_reduce <<<32,           256, 0, stream>>>(part, out);
}